// EmbLoss_67379446940210
// MI455X (gfx1250) — compile-verified
//
#include <hip/hip_runtime.h>
#include <hip/hip_bf16.h>
#include <math.h>

#define N_PIX   (896 * 896)
#define NBATCH  8
#define NBLK    196          // 196 * 4096 == N_PIX exactly
#define TPB     256
#define PPT     16           // pixels per thread
#define NINST   16

typedef __attribute__((ext_vector_type(2))) float v2f;
typedef __attribute__((ext_vector_type(8))) float v8f;

// ---------------------------------------------------------------------------
// Pass 1: kernel-masked segment sums (cnt_k, sum_k) and instance counts cnt_i.
// Per-wave replicated LDS bins (ds_add_f32) -> per-block partial (96 floats).
// ---------------------------------------------------------------------------
__global__ __launch_bounds__(TPB)
void pass1_kernel(const float* __restrict__ emb, const int* __restrict__ instance,
                  const float* __restrict__ kern, const float* __restrict__ tmask,
                  float* __restrict__ part1)
{
    __shared__ float bins[8 * 96];   // 8 waves x [cnt_k(16) | sum_k(64) | cnt_i(16)]
    const int tid = threadIdx.x;
    const int b   = blockIdx.y;
    for (int i = tid; i < 8 * 96; i += TPB) bins[i] = 0.0f;
    __syncthreads();
    float* mb = &bins[(tid >> 5) * 96];

    const size_t nb = (size_t)b * N_PIX;
    const int*   ip = instance + nb;
    const float* kp = kern + nb;
    const float* tp = tmask + nb;
    const float* e0 = emb + (size_t)(b * 4 + 0) * N_PIX;
    const float* e1 = emb + (size_t)(b * 4 + 1) * N_PIX;
    const float* e2 = emb + (size_t)(b * 4 + 2) * N_PIX;
    const float* e3 = emb + (size_t)(b * 4 + 3) * N_PIX;

    const int base = blockIdx.x * (TPB * PPT) + tid;
#pragma unroll
    for (int it = 0; it < PPT; ++it) {
        const int p  = base + it * TPB;
        const int iv = ip[p];
        const bool tm = tp[p] > 0.5f;
        const int ii = tm ? iv : 0;
        if (ii != 0) {
            atomicAdd(&mb[80 + ii], 1.0f);                  // cnt_i
            if (kp[p] > 0.5f) {
                atomicAdd(&mb[ii], 1.0f);                   // cnt_k
                atomicAdd(&mb[16 + ii * 4 + 0], e0[p]);
                atomicAdd(&mb[16 + ii * 4 + 1], e1[p]);
                atomicAdd(&mb[16 + ii * 4 + 2], e2[p]);
                atomicAdd(&mb[16 + ii * 4 + 3], e3[p]);
            }
        }
    }
    __syncthreads();
    for (int c = tid; c < 96; c += TPB) {
        float s = 0.0f;
#pragma unroll
        for (int w = 0; w < 8; ++w) s += bins[w * 96 + c];
        part1[((size_t)b * NBLK + blockIdx.x) * 96 + c] = s;
    }
}

// ---------------------------------------------------------------------------
// Deterministic reduction of pass-1 partials; compute emb_mean (row 0 = 0).
// ---------------------------------------------------------------------------
__global__ __launch_bounds__(128)
void reduce_means_kernel(const float* __restrict__ part1,
                         float* __restrict__ emb_mean,   // [B][16][4]
                         float* __restrict__ cnt_i_tot)  // [B][16]
{
    __shared__ float tot[96];
    const int b = blockIdx.x, c = threadIdx.x;
    if (c < 96) {
        double s = 0.0;
        for (int blk = 0; blk < NBLK; ++blk)
            s += (double)part1[((size_t)b * NBLK + blk) * 96 + c];
        tot[c] = (float)s;
    }
    __syncthreads();
    if (c < 64) {
        const int bin = c >> 2;
        const float m = (bin == 0) ? 0.0f : tot[16 + c] / fmaxf(tot[bin], 1.0f);
        emb_mean[b * 64 + c] = m;
    }
    if (c < 16) cnt_i_tot[b * 16 + c] = tot[80 + c];
}

// ---------------------------------------------------------------------------
// Pass 2: per-pixel pull loss val = log(relu(||e - mean||-0.5)^2 + 1),
// segment-summed per instance (bins 1..15 matter).
// ---------------------------------------------------------------------------
__global__ __launch_bounds__(TPB)
void pass2_kernel(const float* __restrict__ emb, const int* __restrict__ instance,
                  const float* __restrict__ tmask, const float* __restrict__ emb_mean,
                  float* __restrict__ part2)
{
    __shared__ float smean[64];
    __shared__ float bins[8 * 16];
    const int tid = threadIdx.x;
    const int b   = blockIdx.y;
    if (tid < 64) smean[tid] = emb_mean[b * 64 + tid];
    for (int i = tid; i < 8 * 16; i += TPB) bins[i] = 0.0f;
    __syncthreads();
    float* mb = &bins[(tid >> 5) * 16];

    const size_t nb = (size_t)b * N_PIX;
    const int*   ip = instance + nb;
    const float* tp = tmask + nb;
    const float* e0 = emb + (size_t)(b * 4 + 0) * N_PIX;
    const float* e1 = emb + (size_t)(b * 4 + 1) * N_PIX;
    const float* e2 = emb + (size_t)(b * 4 + 2) * N_PIX;
    const float* e3 = emb + (size_t)(b * 4 + 3) * N_PIX;

    const int base = blockIdx.x * (TPB * PPT) + tid;
#pragma unroll
    for (int it = 0; it < PPT; ++it) {
        const int p  = base + it * TPB;
        const int iv = ip[p];
        const bool tm = tp[p] > 0.5f;
        const int ii = tm ? iv : 0;
        if (ii != 0) {
            const float d0 = e0[p] - smean[ii * 4 + 0];
            const float d1 = e1[p] - smean[ii * 4 + 1];
            const float d2 = e2[p] - smean[ii * 4 + 2];
            const float d3 = e3[p] - smean[ii * 4 + 3];
            const float sq = d0 * d0 + d1 * d1 + d2 * d2 + d3 * d3;
            const float dist = sq > 0.0f ? sqrtf(sq) : 0.0f;
            const float rel  = fmaxf(dist - 0.5f, 0.0f);
            atomicAdd(&mb[ii], logf(rel * rel + 1.0f));
        }
    }
    __syncthreads();
    if (tid < 16) {
        float s = 0.0f;
#pragma unroll
        for (int w = 0; w < 8; ++w) s += bins[w * 16 + tid];
        part2[((size_t)b * NBLK + blockIdx.x) * 16 + tid] = s;
    }
}

__global__ __launch_bounds__(32)
void reduce_sumi_kernel(const float* __restrict__ part2, float* __restrict__ sum_i_tot)
{
    const int b = blockIdx.x, c = threadIdx.x;
    if (c < 16) {
        double s = 0.0;
        for (int blk = 0; blk < NBLK; ++blk)
            s += (double)part2[((size_t)b * NBLK + blk) * 16 + c];
        sum_i_tot[b * 16 + c] = (float)s;
    }
}

// ---------------------------------------------------------------------------
// Background select: stable argsort(inst)[:100] == first 100 flat indices with
// inst==0 (instance==0 or training_mask<=0.5). One wave scans with ballot.
// Gathers their embeddings into bg_emb[B][112][4] (rows 100..111 zero pad).
// ---------------------------------------------------------------------------
__global__ __launch_bounds__(32)
void bg_select_kernel(const float* __restrict__ emb, const int* __restrict__ instance,
                      const float* __restrict__ tmask, float* __restrict__ bg_emb)
{
    const int b = blockIdx.x, lane = threadIdx.x;
    float* BG = bg_emb + (size_t)b * 112 * 4;
    for (int i = lane; i < 112 * 4; i += 32) BG[i] = 0.0f;

    const size_t nb = (size_t)b * N_PIX;
    const int*   ip = instance + nb;
    const float* tp = tmask + nb;
    const float* e0 = emb + (size_t)(b * 4) * N_PIX;

    int count = 0;
    for (int base = 0; base < N_PIX && count < 100; base += 32) {
        const int p = base + lane;
        const bool ok = (ip[p] == 0) || !(tp[p] > 0.5f);
        const unsigned mask = (unsigned)__ballot(ok);
        const int prefix = __popc(mask & ((1u << lane) - 1u));
        const int idx = count + prefix;
        if (ok && idx < 100) {
            BG[idx * 4 + 0] = e0[p];
            BG[idx * 4 + 1] = e0[p + N_PIX];
            BG[idx * 4 + 2] = e0[p + 2 * N_PIX];
            BG[idx * 4 + 3] = e0[p + 3 * N_PIX];
        }
        count += __popc(mask);
    }
}

// ---------------------------------------------------------------------------
// Final: one wave. Pairwise push via V_WMMA_F32_16X16X4_F32 Gram of the 16x4
// mean matrix; bg push via 7 more 16x16x4 WMMAs (100 rows padded to 112).
// dmat^2 = n2[i] + n2[j] - 2*G[i,j]. Shuffle-reduce, write scalar mean loss.
// ---------------------------------------------------------------------------
__device__ __forceinline__ float wave_red(float v) {
#pragma unroll
    for (int o = 16; o > 0; o >>= 1) v += __shfl_down(v, o, 32);
    return v;  // valid in lane 0
}

__global__ __launch_bounds__(32)
void final_loss_kernel(const float* __restrict__ emb_mean,   // [B][16][4]
                       const float* __restrict__ cnt_i,      // [B][16]
                       const float* __restrict__ sum_i,      // [B][16]
                       const float* __restrict__ bg_emb,     // [B][112][4]
                       float* __restrict__ out)
{
    __shared__ float s_norm[16];
    __shared__ float s_bgn[16];
    const int lane = threadIdx.x;
    const int r    = lane & 15;
    const bool hi  = lane >= 16;
    float total = 0.0f;

    for (int b = 0; b < NBATCH; ++b) {
        const float* M = emb_mean + b * 64;
        // A (16x4 f32): lanes 0-15 hold (K0,K1) of row r, lanes 16-31 hold (K2,K3).
        // B (4x16) of M^T has the identical per-lane operand -> Gram with one op.
        v2f am;
        am.x = M[r * 4 + (hi ? 2 : 0)];
        am.y = M[r * 4 + (hi ? 3 : 1)];
        if (!hi) {
            float n2 = 0.0f;
#pragma unroll
            for (int f = 0; f < 4; ++f) { const float m = M[r * 4 + f]; n2 += m * m; }
            s_norm[r] = n2;
        }
        __syncthreads();

        const v8f cz = {};
        v8f G = __builtin_amdgcn_wmma_f32_16x16x4_f32(
            false, am, false, am, (short)0, cz, false, false);

        // pairwise push: D VGPR v, lane l -> row v + (l>=16)*8, col l&15
        float pacc = 0.0f;
#pragma unroll
        for (int v = 0; v < 8; ++v) {
            const int m = v + (hi ? 8 : 0);
            const int n = r;
            if (m != n && m != 0 && n != 0) {
                const float sq = s_norm[m] + s_norm[n] - 2.0f * G[v];
                const float d  = sq > 0.0f ? sqrtf(sq) : 0.0f;
                const float rel = fmaxf(3.0f - d, 0.0f);
                pacc += logf(rel * rel + 1.0f);
            }
        }

        // background push: 7 tiles of 16 rows vs the 16 means
        const float* BG = bg_emb + (size_t)b * 112 * 4;
        float bacc = 0.0f;
        for (int t = 0; t < 7; ++t) {
            const int p0 = t * 16;
            v2f ab;
            ab.x = BG[(p0 + r) * 4 + (hi ? 2 : 0)];
            ab.y = BG[(p0 + r) * 4 + (hi ? 3 : 1)];
            __syncthreads();
            if (!hi) {
                float n2 = 0.0f;
#pragma unroll
                for (int f = 0; f < 4; ++f) { const float e = BG[(p0 + r) * 4 + f]; n2 += e * e; }
                s_bgn[r] = n2;
            }
            __syncthreads();
            v8f D = __builtin_amdgcn_wmma_f32_16x16x4_f32(
                false, ab, false, am, (short)0, cz, false, false);
#pragma unroll
            for (int v = 0; v < 8; ++v) {
                const int pl = v + (hi ? 8 : 0);
                const int p  = p0 + pl;
                const int j  = r;
                if (p < 100 && j >= 1) {
                    const float sq = s_bgn[pl] + s_norm[j] - 2.0f * D[v];
                    const float d  = sq > 0.0f ? sqrtf(sq) : 0.0f;
                    const float rel = fmaxf(3.0f - d, 0.0f);
                    bacc += logf(rel * rel + 1.0f);
                }
            }
        }

        // l_agg terms (bins 1..15) and l_reg terms (all 16 bins)
        float aterm = 0.0f, rterm = 0.0f;
        if (!hi) {
            if (r >= 1) aterm = sum_i[b * 16 + r] / fmaxf(cnt_i[b * 16 + r], 1.0f);
            const float n2  = s_norm[r];
            const float nrm = n2 > 0.0f ? sqrtf(n2) : 0.0f;
            rterm = logf(nrm + 1.0f);
        }

        const float psum = wave_red(pacc);
        const float bsum = wave_red(bacc);
        const float asum = wave_red(aterm);
        const float rsum = wave_red(rterm);
        if (lane == 0) {
            const float l_agg = asum / 15.0f;
            const float l_dis = (psum + bsum / 100.0f) / 225.0f;  // 210 pairs + 15 bg cols
            const float l_reg = (rsum / 16.0f) * 0.001f;
            total += l_agg + l_dis + l_reg;
        }
        __syncthreads();   // protect s_norm before next batch overwrites it
    }
    if (lane == 0) out[0] = total / (float)NBATCH;
}

// ---------------------------------------------------------------------------
extern "C" void kernel_launch(void* const* d_in, const int* in_sizes, int n_in,
                              void* d_out, int out_size, void* d_ws, size_t ws_size,
                              hipStream_t stream)
{
    (void)in_sizes; (void)n_in; (void)out_size; (void)ws_size;
    const float* emb      = (const float*)d_in[0];
    const int*   instance = (const int*)d_in[1];
    const float* kern     = (const float*)d_in[2];
    const float* tmask    = (const float*)d_in[3];
    float* out = (float*)d_out;

    float* ws       = (float*)d_ws;
    float* part1    = ws;                                    // 8*196*96 = 150528
    float* part2    = part1 + (size_t)NBATCH * NBLK * 96;    // 8*196*16 = 25088
    float* emb_mean = part2 + (size_t)NBATCH * NBLK * 16;    // 512
    float* cnt_i    = emb_mean + NBATCH * 64;                // 128
    float* sum_i    = cnt_i + NBATCH * 16;                   // 128
    float* bg       = sum_i + NBATCH * 16;                   // 8*112*4 = 3584

    pass1_kernel<<<dim3(NBLK, NBATCH), TPB, 0, stream>>>(emb, instance, kern, tmask, part1);
    reduce_means_kernel<<<NBATCH, 128, 0, stream>>>(part1, emb_mean, cnt_i);
    pass2_kernel<<<dim3(NBLK, NBATCH), TPB, 0, stream>>>(emb, instance, tmask, emb_mean, part2);
    reduce_sumi_kernel<<<NBATCH, 32, 0, stream>>>(part2, sum_i);
    bg_select_kernel<<<NBATCH, 32, 0, stream>>>(emb, instance, tmask, bg);
    final_loss_kernel<<<1, 32, 0, stream>>>(emb_mean, cnt_i, sum_i, bg, out);
}